// ComputeLayer_15951508538131
// MI455X (gfx1250) — compile-verified
//
#include <hip/hip_runtime.h>
#include <math.h>
#include <stdint.h>

typedef __attribute__((ext_vector_type(2))) float v2f;
typedef __attribute__((ext_vector_type(8))) float v8f;
typedef __attribute__((ext_vector_type(4))) unsigned int v4u;
typedef __attribute__((ext_vector_type(8))) int v8i;
typedef __attribute__((ext_vector_type(4))) int v4i;

#define D_MODEL 512
#define HDIM    128
#define NPROG   10
#define MAXO    4
#define MAXR    3
#define BM      64                  // rows per block (hidden tile stays resident)
#define KC      64                  // W1 K-chunk staged in LDS
#define LDW     136                 // W1 tile LDS stride (floats): TDM pad 128+8
#define HHALF   258                 // hidden half-row stride: TDM pad 256+2
#define HROW    (2 * HHALF)         // 516 floats per hidden row; 516 mod 64 = 4
#define LDG     132                 // gelu tile stride

#define HID_F   (BM * HROW)                 // 33024 floats (hidden tile)
#define WSZ     (KC * LDW)                  // 8704 floats per W1 buffer
#define W_OFF   HID_F
#define G_OFF   (W_OFF + 2 * WSZ)           // 50432
#define OPS_OFF (G_OFF + BM * LDG)          // 58880
#define TOTAL_F (OPS_OFF + BM * MAXO)       // 59136 floats
#define SMEM_BYTES (TOTAL_F * 4)            // 236544 B  (< 320KB WGP LDS)

#define NCHUNK  (D_MODEL / KC)              // 8
#define NG      (NPROG * NCHUNK)            // 80 pipelined W1 chunk loads

__device__ __forceinline__ float gelu_exact(float x) {
    return 0.5f * x * (1.0f + erff(x * 0.70710678118654752f));
}

// ---- Tensor Data Mover: 2D f32 tile global -> LDS (D# per ISA ch.8.3/8.4) ----
__device__ __forceinline__ void tdm_load_2d(unsigned int lds_byte_off,
                                            const float* gaddr,
                                            unsigned int tile_d0, unsigned int tile_d1,
                                            unsigned int tens_d0,
                                            unsigned int stride0,
                                            unsigned int pad_interval_code,
                                            unsigned int pad_amount_code)
{
    unsigned long long ga = (unsigned long long)(uintptr_t)gaddr;
    v4u g0;
    g0.x = 1u;                                             // count=1, no gather
    g0.y = lds_byte_off;                                   // lds_addr (bytes)
    g0.z = (unsigned int)(ga & 0xffffffffull);             // global_addr[31:0]
    g0.w = (unsigned int)((ga >> 32) & 0x01ffffffull)      // global_addr[56:32]
         | (2u << 30);                                     // type=2 (image)
    v8i g1;
    g1[0] = (int)((2u << 16)                               // data_size = 4B
                | (1u << 20)                               // pad_enable
                | (pad_interval_code << 22)
                | (pad_amount_code << 25));
    g1[1] = (int)((tens_d0 & 0xffffu) << 16);              // tensor_dim0[15:0]
    g1[2] = (int)(((tens_d0 >> 16) & 0xffffu)
                | ((tile_d1 & 0xffffu) << 16));            // tensor_dim1[15:0]
    g1[3] = (int)((tile_d0 & 0xffffu) << 16);              // tile_dim0
    g1[4] = (int)(tile_d1 & 0xffffu);                      // tile_dim1 | tile_dim2=0
    g1[5] = (int)stride0;                                  // tensor_dim0_stride[31:0]
    g1[6] = 0;
    g1[7] = 0;
    v4i z4 = {};
    v8i z8 = {};
    __builtin_amdgcn_tensor_load_to_lds(g0, g1, z4, z4, z8, 0);
}

__global__ __launch_bounds__(256)
void fused_program_kernel(const float* __restrict__ hidden,
                          const int*   __restrict__ pids,
                          const float* __restrict__ W1,
                          const float* __restrict__ b1,
                          const float* __restrict__ W2,
                          const float* __restrict__ b2,
                          float* __restrict__ out_res,
                          float* __restrict__ out_ops,
                          int B)
{
    extern __shared__ float smem[];
    const unsigned int ldsBase = (unsigned int)(uintptr_t)(void*)smem;

    const int rowBase = blockIdx.x * BM;
    const int tid     = threadIdx.x;
    const int lane    = tid & 31;
    const int wave    = tid >> 5;

    const int wRow = (wave & 3) * 16;
    const int wCol = (wave >> 2) * 64;
    const int lm   = lane & 15;
    const int lh   = (lane >> 4) << 1;

    // prologue: hidden tile (once) + first W1 chunk
    if (wave == 0) {
        // 64 x 512 tile, pad 2 DWORDs every 256 -> half-row stride 258
        tdm_load_2d(ldsBase, &hidden[(size_t)rowBase * D_MODEL],
                    D_MODEL, BM, D_MODEL, D_MODEL, 7, 1);
        // W1[p=0] chunk 0: 64 x 128, pad 8 every 128 -> stride 136
        tdm_load_2d(ldsBase + W_OFF * 4, W1, HDIM, KC, HDIM, HDIM, 6, 7);
    }

    v8f acc[4] = {};
    const v8f vzero = {};

    for (int g = 0; g < NG; ++g) {
        const int p = g >> 3;
        const int c = g & 7;
        if (c == 0) { acc[0] = vzero; acc[1] = vzero; acc[2] = vzero; acc[3] = vzero; }

        if (wave == 0) {
            if (g + 1 < NG) {   // prefetch next W1 chunk into other buffer
                const int pn = (g + 1) >> 3, cn = (g + 1) & 7;
                tdm_load_2d(ldsBase + (W_OFF + ((g + 1) & 1) * WSZ) * 4,
                            &W1[((size_t)pn * D_MODEL + cn * KC) * HDIM],
                            HDIM, KC, HDIM, HDIM, 6, 7);
                __builtin_amdgcn_s_wait_tensorcnt(1);  // chunk g (and hidden) landed
            } else {
                __builtin_amdgcn_s_wait_tensorcnt(0);
            }
        }
        __syncthreads();

        const float* ldsW = smem + W_OFF + (g & 1) * WSZ;      // [KC][LDW]
        // A base inside resident hidden tile: chunk c -> half (c>>2), quarter (c&3)
        const int hBase = (wRow + lm) * HROW + (c >> 2) * HHALF + (c & 3) * KC + lh;

        #pragma unroll
        for (int kk = 0; kk < KC; kk += 4) {
            v2f a = *(const v2f*)&smem[hBase + kk];
            #pragma unroll
            for (int n = 0; n < 4; ++n) {
                int col = wCol + n * 16 + lm;
                v2f b;
                b.x = ldsW[(kk + lh)     * LDW + col];
                b.y = ldsW[(kk + lh + 1) * LDW + col];
                acc[n] = __builtin_amdgcn_wmma_f32_16x16x4_f32(
                    false, a, false, b, (short)0, acc[n], false, false);
            }
        }
        __syncthreads();   // W buffer reads done before TDM refills it

        if (c == 7) {
            // ---- epilogue for program p ----
            float* ldsG   = smem + G_OFF;     // [BM][LDG]
            float* ldsOps = smem + OPS_OFF;   // [BM][4]
            #pragma unroll
            for (int n = 0; n < 4; ++n) {
                int   col  = wCol + n * 16 + lm;
                float bias = b1[p * HDIM + col];
                #pragma unroll
                for (int r = 0; r < 8; ++r) {
                    int row = wRow + r + ((lane >> 4) << 3);
                    ldsG[row * LDG + col] = gelu_exact(acc[n][r] + bias);
                }
            }
            __syncthreads();

            {   // second GEMM (128 -> 4), one scalar per thread
                int   row = tid >> 2;
                int   o   = tid & 3;
                float s   = b2[p * MAXO + o];
                #pragma unroll 8
                for (int h = 0; h < HDIM; ++h)
                    s += ldsG[row * LDG + h] * W2[((size_t)p * HDIM + h) * MAXO + o];
                int gRow = rowBase + row;
                out_ops[((size_t)p * B + gRow) * MAXO + o] = s;   // pre-round ops
                ldsOps[row * 4 + o] = rintf(s);                   // round half-even
            }
            __syncthreads();

            if (tid < BM) {
                int grow = rowBase + tid;
                if (pids[grow] == p) {
                    float a = ldsOps[tid * 4 + 0], b = ldsOps[tid * 4 + 1];
                    float cc = ldsOps[tid * 4 + 2], d = ldsOps[tid * 4 + 3];
                    float o0 = 0.f, o1 = 0.f, o2 = 0.f;
                    switch (p) {
                        case 0: o0 = a + b; break;
                        case 1: o0 = a - b; break;
                        case 2: o0 = a * b; break;
                        case 3: { float sb = (b == 0.f) ? 1.f : b;
                                  o0 = a - floorf(a / sb) * sb; } break;
                        case 4: { float sb = (b == 0.f) ? 1.f : b;
                                  o0 = floorf(a / sb); } break;
                        case 5: case 6: case 7: {
                            float t = a * 60.f + b + ((p == 5) ? (cc * 60.f + d)
                                                               : -(cc * 60.f + d));
                            if (p == 7) t = fabsf(t);
                            float days = floorf(t / 1440.f);
                            float rem  = t - days * 1440.f;
                            float hh   = floorf(rem / 60.f);
                            o0 = hh; o1 = rem - hh * 60.f; o2 = days;
                        } break;
                        case 8: o0 = (a > b)  ? 1.f : 0.f; break;
                        case 9: o0 = (a == b) ? 1.f : 0.f; break;
                    }
                    out_res[(size_t)grow * MAXR + 0] = o0;
                    out_res[(size_t)grow * MAXR + 1] = o1;
                    out_res[(size_t)grow * MAXR + 2] = o2;
                }
            }
            // next iteration starts with a __syncthreads() before any LDS reuse
        }
    }
}

__global__ __launch_bounds__(256)
void router_kernel(const float* __restrict__ hidden,
                   const float* __restrict__ Wr,
                   const float* __restrict__ br,
                   const int*   __restrict__ pids,
                   float* __restrict__ out_router,
                   float* __restrict__ out_pid,
                   int B)
{
    int wave = threadIdx.x >> 5, lane = threadIdx.x & 31;
    int row  = blockIdx.x * 8 + wave;
    if (row >= B) return;

    float acc[NPROG] = {};
    for (int k = lane; k < D_MODEL; k += 32) {
        float hv = hidden[(size_t)row * D_MODEL + k];
        #pragma unroll
        for (int j = 0; j < NPROG; ++j)
            acc[j] += hv * Wr[k * NPROG + j];
    }
    #pragma unroll
    for (int j = 0; j < NPROG; ++j)
        #pragma unroll
        for (int off = 16; off > 0; off >>= 1)
            acc[j] += __shfl_xor(acc[j], off, 32);

    if (lane < NPROG)
        out_router[(size_t)row * NPROG + lane] = acc[lane] + br[lane];
    if (lane == 0)
        out_pid[row] = (float)pids[row];
}

extern "C" void kernel_launch(void* const* d_in, const int* in_sizes, int n_in,
                              void* d_out, int out_size, void* d_ws, size_t ws_size,
                              hipStream_t stream) {
    const float* hidden = (const float*)d_in[0];
    const int*   pids   = (const int*)  d_in[1];
    const float* Wr     = (const float*)d_in[2];
    const float* br     = (const float*)d_in[3];
    const float* W1     = (const float*)d_in[4];
    const float* b1     = (const float*)d_in[5];
    const float* W2     = (const float*)d_in[6];
    const float* b2     = (const float*)d_in[7];

    const int B = in_sizes[1];

    float* out        = (float*)d_out;  // concat: results | router | ops | pids
    float* out_res    = out;
    float* out_router = out + (size_t)B * MAXR;
    float* out_ops    = out + (size_t)B * (MAXR + NPROG);
    float* out_pid    = out_ops + (size_t)NPROG * B * MAXO;

    fused_program_kernel<<<B / BM, 256, SMEM_BYTES, stream>>>(hidden, pids, W1, b1, W2, b2,
                                                              out_res, out_ops, B);
    router_kernel<<<(B + 7) / 8, 256, 0, stream>>>(hidden, Wr, br, pids,
                                                   out_router, out_pid, B);
}